// TransactionLayer_22754736734795
// MI455X (gfx1250) — compile-verified
//
#include <hip/hip_runtime.h>
#include <float.h>

// ---------------------------------------------------------------------------
// Global max pooling over spatial dims: in [B, H, W, C] f32 -> out [B, C] f32
// B=64, H=W=56 (HW=3136), C=256.  Pure HBM-bound streaming reduction.
// ---------------------------------------------------------------------------

typedef __attribute__((ext_vector_type(4))) float v4f;

#define BATCH    64
#define SPATIAL  3136          // 56*56
#define CH       256
#define CG       64            // CH/4 float4 channel-groups
#define CHUNKS   14            // 3136 = 14 * 224
#define SPP      (SPATIAL / CHUNKS)   // 224 spatial positions per chunk

__device__ __forceinline__ v4f vmax4(v4f a, v4f b) {
    v4f r;
    r.x = fmaxf(a.x, b.x);
    r.y = fmaxf(a.y, b.y);
    r.z = fmaxf(a.z, b.z);
    r.w = fmaxf(a.w, b.w);
    return r;
}

// Phase A: one block per (batch, spatial-chunk). 256 threads:
//   cg = tid & 63  -> float4 channel group (coalesced b128 across a wave)
//   sp = tid >> 6  -> 4 spatial rows in flight
// Writes per-chunk channel maxima to ws[b*CHUNKS + chunk][256].
__global__ __launch_bounds__(256) void gmax_partial(const float* __restrict__ in,
                                                    float* __restrict__ ws)
{
    const int blk = blockIdx.x;                  // 0 .. BATCH*CHUNKS-1
    const int b   = blk / CHUNKS;
    const int ck  = blk - b * CHUNKS;
    const int tid = threadIdx.x;
    const int cg  = tid & (CG - 1);
    const int sp  = tid >> 6;                    // 0..3

    const v4f* base = (const v4f*)in
                    + ((size_t)b * SPATIAL + (size_t)ck * SPP) * CG + cg;
    // stride between consecutive spatial positions = CG v4f elements (256 floats)

    v4f m = (v4f){-FLT_MAX, -FLT_MAX, -FLT_MAX, -FLT_MAX};

    // 14 outer iterations, 4 independent 128-bit non-temporal loads each.
    for (int p = sp; p < SPP; p += 16) {
        __builtin_prefetch(base + (size_t)(p + 16) * CG, 0, 0); // global_prefetch_b8 (speculative)
        v4f v0 = __builtin_nontemporal_load(base + (size_t)(p     ) * CG);
        v4f v1 = __builtin_nontemporal_load(base + (size_t)(p +  4) * CG);
        v4f v2 = __builtin_nontemporal_load(base + (size_t)(p +  8) * CG);
        v4f v3 = __builtin_nontemporal_load(base + (size_t)(p + 12) * CG);
        m = vmax4(m, vmax4(vmax4(v0, v1), vmax4(v2, v3)));
    }

    // Fold the 4 spatial lanes per channel-group through LDS (ds_*_b128).
    __shared__ v4f sm[256];
    sm[tid] = m;
    __syncthreads();
    if (sp == 0) {
        v4f r = vmax4(vmax4(sm[cg], sm[cg + 64]),
                      vmax4(sm[cg + 128], sm[cg + 192]));
        ((v4f*)ws)[(size_t)blk * CG + cg] = r;
    }
}

// Phase B: one block per batch image; lane cg folds the 14 chunk partials.
__global__ __launch_bounds__(64) void gmax_final(const float* __restrict__ ws,
                                                 float* __restrict__ out)
{
    const int b  = blockIdx.x;
    const int cg = threadIdx.x;                  // 0..63
    const v4f* w = (const v4f*)ws + (size_t)b * CHUNKS * CG + cg;
    v4f m = w[0];
#pragma unroll
    for (int k = 1; k < CHUNKS; ++k) m = vmax4(m, w[(size_t)k * CG]);
    ((v4f*)out)[(size_t)b * CG + cg] = m;
}

// Fallback (only if workspace is too small): one block reduces a whole image.
__global__ __launch_bounds__(256) void gmax_full(const float* __restrict__ in,
                                                 float* __restrict__ out)
{
    const int b   = blockIdx.x;
    const int tid = threadIdx.x;
    const int cg  = tid & (CG - 1);
    const int sp  = tid >> 6;

    const v4f* base = (const v4f*)in + (size_t)b * SPATIAL * CG + cg;

    v4f m = (v4f){-FLT_MAX, -FLT_MAX, -FLT_MAX, -FLT_MAX};
    for (int p = sp; p < SPATIAL; p += 16) {
        v4f v0 = __builtin_nontemporal_load(base + (size_t)(p     ) * CG);
        v4f v1 = __builtin_nontemporal_load(base + (size_t)(p +  4) * CG);
        v4f v2 = __builtin_nontemporal_load(base + (size_t)(p +  8) * CG);
        v4f v3 = __builtin_nontemporal_load(base + (size_t)(p + 12) * CG);
        m = vmax4(m, vmax4(vmax4(v0, v1), vmax4(v2, v3)));
    }

    __shared__ v4f sm[256];
    sm[tid] = m;
    __syncthreads();
    if (sp == 0) {
        v4f r = vmax4(vmax4(sm[cg], sm[cg + 64]),
                      vmax4(sm[cg + 128], sm[cg + 192]));
        ((v4f*)out)[(size_t)b * CG + cg] = r;
    }
}

extern "C" void kernel_launch(void* const* d_in, const int* in_sizes, int n_in,
                              void* d_out, int out_size, void* d_ws, size_t ws_size,
                              hipStream_t stream)
{
    (void)in_sizes; (void)n_in; (void)out_size;
    const float* in  = (const float*)d_in[0];
    float*       out = (float*)d_out;

    const size_t need = (size_t)BATCH * CHUNKS * CH * sizeof(float); // ~896 KB
    if (d_ws && ws_size >= need) {
        float* ws = (float*)d_ws;
        gmax_partial<<<BATCH * CHUNKS, 256, 0, stream>>>(in, ws);
        gmax_final<<<BATCH, 64, 0, stream>>>(ws, out);
    } else {
        gmax_full<<<BATCH, 256, 0, stream>>>(in, out);
    }
}